// EdgeToVertex_10909216932338
// MI455X (gfx1250) — compile-verified
//
#include <hip/hip_runtime.h>

// D = A(16x4) * B(4x16) + C, f32 — used as a column-sum engine with A = ones:
// D[m,n] = sum_k B[k,n] + C[m,n], so only the lane->column mapping of B matters
// (N = lane%16 in all documented layouts); row-to-K-slot assignment is arbitrary.
typedef float v2f __attribute__((ext_vector_type(2)));
typedef float v8f __attribute__((ext_vector_type(8)));

#define N   256
#define NT  16   // 16x16 tiles per dimension

// Issue all 8 loads of a 16x16 tile (this lane's column, 8 row slots) at once.
__device__ __forceinline__ void load_tile(const float* __restrict__ T,
                                          int c, int half, float e[8]) {
    #pragma unroll
    for (int q = 0; q < 4; ++q) {
        e[2 * q]     = T[(4 * q + half) * N + c];
        e[2 * q + 1] = T[(4 * q + half + 2) * N + c];
    }
}

// h >= w mask for the diagonal tile (tile-local rows vs column c).
__device__ __forceinline__ void mask_diag(int c, int half, float e[8]) {
    #pragma unroll
    for (int q = 0; q < 4; ++q) {
        e[2 * q]     = (4 * q + half     >= c) ? e[2 * q]     : 0.0f;
        e[2 * q + 1] = (4 * q + half + 2 >= c) ? e[2 * q + 1] : 0.0f;
    }
}

__device__ __forceinline__ void wmma_tile(const float e[8], v2f ones, v8f& acc) {
    #pragma unroll
    for (int q = 0; q < 4; ++q) {
        v2f bm; bm[0] = e[2 * q]; bm[1] = e[2 * q + 1];
        acc = __builtin_amdgcn_wmma_f32_16x16x4_f32(
            false, ones, false, bm, (short)0, acc, false, false);
    }
}

// Tril column sums of column-tile j: row tiles i = j..15, double-buffered so the
// next tile's 8 loads are in flight while the current tile feeds the WMMA chain.
__device__ __forceinline__ void colsum_tile_stripe(
    const float* __restrict__ Xb, int j, int c, int half, v2f ones, v8f& acc) {
    const int col0 = j * 16;
    const float* T = Xb + (size_t)(j * 16) * N + col0;

    float cur[8];
    load_tile(T, c, half, cur);   // diagonal tile first
    mask_diag(c, half, cur);

    for (int i = j + 1; i < NT; ++i) {
        float nxt[8];
        load_tile(Xb + (size_t)(i * 16) * N + col0, c, half, nxt);  // prefetch i
        wmma_tile(cur, ones, acc);                                  // consume i-1
        #pragma unroll
        for (int t = 0; t < 8; ++t) cur[t] = nxt[t];
    }
    wmma_tile(cur, ones, acc);
}

__global__ __launch_bounds__(256)
void tril_cross_conv_kernel(const float* __restrict__ x, float* __restrict__ out) {
    __shared__ float S[N];       // S_col(w): tril column sums
    __shared__ float Rrow[129];  // x[b, 128, 0..128]

    const int b     = blockIdx.x;
    const float* Xb = x + (size_t)b * (N * N);

    const int tid  = threadIdx.x;
    const int wave = tid >> 5;     // 0..7
    const int lane = tid & 31;
    const int c    = lane & 15;    // tile-local column (B layout: N = lane%16)
    const int half = lane >> 4;    // 0 or 1

    // Balanced pairing: wave w owns column tiles j1=w (16-w row tiles) and
    // j2=15-w (w+1 row tiles) -> exactly 17 row tiles per wave.
    const int j1 = wave;
    const int j2 = (NT - 1) - wave;

    v8f acc1 = {0.f, 0.f, 0.f, 0.f, 0.f, 0.f, 0.f, 0.f};
    v8f acc2 = {0.f, 0.f, 0.f, 0.f, 0.f, 0.f, 0.f, 0.f};
    v2f ones = {1.0f, 1.0f};       // A = all ones

    colsum_tile_stripe(Xb, j1, c, half, ones, acc1);
    colsum_tile_stripe(Xb, j2, c, half, ones, acc2);

    // All D rows are identical; first VGPR, lanes 0..15 hold colsum(N = lane).
    if (lane < 16) {
        S[j1 * 16 + lane] = acc1[0];
        S[j2 * 16 + lane] = acc2[0];
    }

    // Stage row 128 (tril keeps only cols 0..128 of it).
    if (tid < 129) Rrow[tid] = Xb[128 * N + tid];

    __syncthreads();

    {
        const int w  = tid;
        const int lo = (w >= 129) ? (w - 128) : 0;
        float rs = 0.0f;
        for (int cc = lo; cc <= 128; ++cc) rs += Rrow[cc];   // windowed row-128 sum
        float v = S[w] + rs;
        if (w <= 128) v -= 2.0f * Rrow[w];  // (128,w) excluded from both col & row terms
        out[(size_t)b * N + w] = v;
    }
}

extern "C" void kernel_launch(void* const* d_in, const int* in_sizes, int n_in,
                              void* d_out, int out_size, void* d_ws, size_t ws_size,
                              hipStream_t stream) {
    (void)in_sizes; (void)n_in; (void)out_size; (void)d_ws; (void)ws_size;
    const float* x = (const float*)d_in[0];   // [512, 256, 256, 1] f32
    float* out     = (float*)d_out;           // [512, 1, 256, 1] f32
    tril_cross_conv_kernel<<<dim3(512), dim3(256), 0, stream>>>(x, out);
}